// GNN_29446295781863
// MI455X (gfx1250) — compile-verified
//
#include <hip/hip_runtime.h>
#include <hip/hip_bf16.h>
#include <math.h>

#define N_NODES  100000
#define N_EDGES  1600000
#define N_GRAPHS 128
#define HDIM     128

typedef __attribute__((ext_vector_type(2))) float v2f;
typedef __attribute__((ext_vector_type(8))) float v8f;

// ---------------------------------------------------------------------------
// Utility: zero a float buffer
// ---------------------------------------------------------------------------
__global__ void zero_f32(float* __restrict__ p, int n) {
  int i = blockIdx.x * 256 + threadIdx.x;
  if (i < n) p[i] = 0.0f;
}

// ---------------------------------------------------------------------------
// Edge scatter for 4-wide features (layer 0 aggregation)
// agg[dst] += x[src]  (4 floats per edge)
// ---------------------------------------------------------------------------
__global__ void scatter_edges4(const int* __restrict__ src, const int* __restrict__ dst,
                               const float* __restrict__ x, float* __restrict__ agg, int nE) {
  int e = blockIdx.x * 256 + threadIdx.x;
  if (e >= nE) return;
  int s = src[e], d = dst[e];
  const float* xs = x + (size_t)s * 4;
  float* ad = agg + (size_t)d * 4;
  atomicAdd(ad + 0, xs[0]);
  atomicAdd(ad + 1, xs[1]);
  atomicAdd(ad + 2, xs[2]);
  atomicAdd(ad + 3, xs[3]);
}

// ---------------------------------------------------------------------------
// Edge scatter for 128-wide features: 128 consecutive threads handle one edge.
// agg[dst][f] += h[src][f].  L2-resident (51MB buffers << 192MB L2).
// ---------------------------------------------------------------------------
__global__ void scatter_edges128(const int* __restrict__ src, const int* __restrict__ dst,
                                 const float* __restrict__ h, float* __restrict__ agg, int nE) {
  long long g = (long long)blockIdx.x * 256 + threadIdx.x;
  int e = (int)(g >> 7);
  int f = (int)(g & 127);
  if (e >= nE) return;
  atomicAdd(&agg[(size_t)dst[e] * HDIM + f], h[(size_t)src[e] * HDIM + f]);
}

// ---------------------------------------------------------------------------
// Fused dual-GEMM + bias + relu using fp32 WMMA (V_WMMA_F32_16X16X4_F32):
//   Y[m][n] = relu( sum_k A1[m][k]*W1[k][n] + sum_k A2[m][k]*W2[k][n] + bias[n] )
// N fixed to 128; K is 4 (layer 0) or 128; M must be a multiple of 16.
// Block = 256 threads = 8 waves; block covers 128 rows, each wave a 16x128 tile.
//
// W is staged into LDS in PAIR-INTERLEAVED layout so each B fragment is one
// aligned 8-byte ds_load_b64 straight into an even VGPR pair (no mov shuffles):
//   lds[(k/2)*256 + n*2 + (k&1)] = W[k][n]
//
// Fragment layouts per CDNA5 ISA 7.12.2 (wave32):
//   A 16x4 f32 : lane&15 -> M ; VGPR0 = K {0|2}, VGPR1 = K {1|3} (hi half +2)
//   B  4x16 f32: lane&15 -> N ; VGPR0 = K {0|2}, VGPR1 = K {1|3}
//   C/D 16x16  : VGPR v -> M = v + 8*(lane>=16) ; lane&15 -> N
// ---------------------------------------------------------------------------
__global__ void fused_gemm2_bias_relu(const float* __restrict__ A1, const float* __restrict__ W1,
                                      const float* __restrict__ A2, const float* __restrict__ W2,
                                      const float* __restrict__ bias, float* __restrict__ Y,
                                      int M, int K) {
  extern __shared__ float ldsW[];          // K * 128 floats, pair-interleaved
  const int tid  = threadIdx.x;            // 0..255
  const int wave = tid >> 5;               // 0..7
  const int lane = tid & 31;
  const int l16  = lane & 15;
  const int hi   = lane >> 4;              // 0 or 1
  const int tileRow = blockIdx.x * 128 + wave * 16;

  v8f c[8] = {};                           // 16 x 128 accumulator tile per wave

  const float* Aarr[2] = {A1, A2};
  const float* Warr[2] = {W1, W2};

  for (int pass = 0; pass < 2; ++pass) {
    const float* W = Warr[pass];
    if (W == nullptr) break;
    const float* A = Aarr[pass];

    // Stage W (K x 128) into LDS, pair-interleaving adjacent k rows:
    // pair p = (kp, n): lds[p*2 .. p*2+1] = { W[2*kp][n], W[2*kp+1][n] }
    const int nPairs = (K >> 1) * HDIM;
    for (int p = tid; p < nPairs; p += 256) {
      int kp = p >> 7;          // pair row
      int n  = p & 127;
      v2f v;
      v.x = W[(size_t)(2 * kp) * HDIM + n];
      v.y = W[(size_t)(2 * kp + 1) * HDIM + n];
      *(v2f*)(ldsW + (size_t)p * 2) = v;   // 8B-aligned ds_store_b64
    }
    __syncthreads();

    if (tileRow < M) {
      const float* Arow = A + (size_t)(tileRow + l16) * K;
      for (int k0 = 0; k0 < K; k0 += 4) {
        // A fragment: aligned 8B load of (K = k0+2*hi, k0+2*hi+1)
        v2f a = *(const v2f*)(Arow + k0 + 2 * hi);
        // B fragments: pair row kp = k0/2 + hi, column nt*16 + l16
        const float* bp = ldsW + ((size_t)((k0 >> 1) + hi) * HDIM + l16) * 2;
#pragma unroll
        for (int nt = 0; nt < 8; ++nt) {
          v2f b = *(const v2f*)(bp + nt * 32);   // aligned ds_load_b64
          c[nt] = __builtin_amdgcn_wmma_f32_16x16x4_f32(
              /*neg_a=*/false, a, /*neg_b=*/false, b,
              /*c_mod=*/(short)0, c[nt], /*reuse_a=*/false, /*reuse_b=*/false);
        }
      }
    }
    __syncthreads();  // before re-staging the second weight matrix
  }

  if (tileRow < M) {
#pragma unroll
    for (int nt = 0; nt < 8; ++nt) {
      int col = nt * 16 + l16;
      float bv = bias ? bias[col] : 0.0f;
#pragma unroll
      for (int v = 0; v < 8; ++v) {
        int row = tileRow + v + 8 * hi;
        float val = c[nt][v] + bv;
        Y[(size_t)row * HDIM + col] = val > 0.0f ? val : 0.0f;
      }
    }
  }
}

// ---------------------------------------------------------------------------
// Global mean-pool: accumulate per-graph sums and counts
// ---------------------------------------------------------------------------
__global__ void pool_accum(const int* __restrict__ batch, const float* __restrict__ h,
                           float* __restrict__ sums, float* __restrict__ cnt, int nNodes) {
  long long gi = (long long)blockIdx.x * 256 + threadIdx.x;
  int i = (int)(gi >> 7);
  int f = (int)(gi & 127);
  if (i >= nNodes) return;
  int g = batch[i];
  atomicAdd(&sums[(size_t)g * HDIM + f], h[(size_t)i * HDIM + f]);
  if (f == 0) atomicAdd(&cnt[g], 1.0f);
}

__global__ void pool_div(const float* __restrict__ sums, const float* __restrict__ cnt,
                         float* __restrict__ g) {
  int idx = blockIdx.x * 256 + threadIdx.x;   // 128*128 elements
  if (idx >= N_GRAPHS * HDIM) return;
  float c = cnt[idx >> 7];
  g[idx] = sums[idx] / fmaxf(c, 1.0f);
}

// ---------------------------------------------------------------------------
// Final 128x128 @ 128x2 + bias + sigmoid  (tiny: single block, 256 threads)
// ---------------------------------------------------------------------------
__global__ void final_sigmoid(const float* __restrict__ g, const float* __restrict__ Wl,
                              const float* __restrict__ bl, float* __restrict__ out) {
  int idx = threadIdx.x;                      // 0..255 -> (graph, out)
  if (idx >= N_GRAPHS * 2) return;
  int gr = idx >> 1;
  int o  = idx & 1;
  float acc = bl[o];
#pragma unroll 8
  for (int k = 0; k < HDIM; ++k) acc += g[(size_t)gr * HDIM + k] * Wl[k * 2 + o];
  out[idx] = 1.0f / (1.0f + __expf(-acc));
}

// ---------------------------------------------------------------------------
// Launcher
// ---------------------------------------------------------------------------
extern "C" void kernel_launch(void* const* d_in, const int* in_sizes, int n_in,
                              void* d_out, int out_size, void* d_ws, size_t ws_size,
                              hipStream_t stream) {
  (void)in_sizes; (void)n_in; (void)out_size; (void)ws_size;

  const float* x     = (const float*)d_in[0];   // (100000, 4)
  const int*   ei    = (const int*)  d_in[1];   // (2, 1600000) src row 0, dst row 1
  const int*   batch = (const int*)  d_in[2];   // (100000,)
  const float* Wr0 = (const float*)d_in[3];  const float* br0 = (const float*)d_in[4];  const float* Wo0 = (const float*)d_in[5];
  const float* Wr1 = (const float*)d_in[6];  const float* br1 = (const float*)d_in[7];  const float* Wo1 = (const float*)d_in[8];
  const float* Wr2 = (const float*)d_in[9];  const float* br2 = (const float*)d_in[10]; const float* Wo2 = (const float*)d_in[11];
  const float* Wr3 = (const float*)d_in[12]; const float* br3 = (const float*)d_in[13]; const float* Wo3 = (const float*)d_in[14];
  const float* W5 = (const float*)d_in[15]; const float* b5 = (const float*)d_in[16];
  const float* W6 = (const float*)d_in[17]; const float* b6 = (const float*)d_in[18];
  const float* W7 = (const float*)d_in[19]; const float* b7 = (const float*)d_in[20];
  const float* W8 = (const float*)d_in[21]; const float* b8 = (const float*)d_in[22];
  const float* Wl = (const float*)d_in[23]; const float* bl = (const float*)d_in[24];

  const int* src = ei;
  const int* dst = ei + N_EDGES;

  // Carve scratch from d_ws (256B aligned slices); total ~156 MB.
  size_t off = 0;
  auto carve = [&](size_t bytes) -> float* {
    float* p = (float*)((char*)d_ws + off);
    off += (bytes + 255) & ~(size_t)255;
    return p;
  };
  float* agg0 = carve((size_t)N_NODES * 4 * sizeof(float));        // 1.6 MB
  float* hA   = carve((size_t)N_NODES * HDIM * sizeof(float));     // 51.2 MB
  float* hB   = carve((size_t)N_NODES * HDIM * sizeof(float));     // 51.2 MB
  float* agg  = carve((size_t)N_NODES * HDIM * sizeof(float));     // 51.2 MB
  float* sums = carve((size_t)(N_GRAPHS * HDIM + N_GRAPHS) * sizeof(float)); // sums + cnt contiguous
  float* cnt  = sums + N_GRAPHS * HDIM;
  float* g0   = carve((size_t)N_GRAPHS * HDIM * sizeof(float));
  float* g1   = carve((size_t)N_GRAPHS * HDIM * sizeof(float));

  const int gemmGrid = (N_NODES + 127) / 128;   // 782 blocks, M=100000 divisible by 16
  const size_t lds4   = (size_t)4 * HDIM * sizeof(float);     // 2 KB
  const size_t lds128 = (size_t)HDIM * HDIM * sizeof(float);  // 64 KB

  // ---- Layer 0 (F_IN=4 -> H=128) ----
  zero_f32<<<(N_NODES * 4 + 255) / 256, 256, 0, stream>>>(agg0, N_NODES * 4);
  scatter_edges4<<<(N_EDGES + 255) / 256, 256, 0, stream>>>(src, dst, x, agg0, N_EDGES);
  fused_gemm2_bias_relu<<<gemmGrid, 256, lds4, stream>>>(agg0, Wr0, x, Wo0, br0, hA, N_NODES, 4);

  // ---- Layers 1..3 (H -> H) ----
  const float* Wr[3] = {Wr1, Wr2, Wr3};
  const float* br[3] = {br1, br2, br3};
  const float* Wo[3] = {Wo1, Wo2, Wo3};
  float* cur = hA;
  float* nxt = hB;
  const long long scatterThreads = (long long)N_EDGES * HDIM;   // 204.8M
  const int scatterGrid = (int)((scatterThreads + 255) / 256);  // 800000
  for (int L = 0; L < 3; ++L) {
    zero_f32<<<(N_NODES * HDIM + 255) / 256, 256, 0, stream>>>(agg, N_NODES * HDIM);
    scatter_edges128<<<scatterGrid, 256, 0, stream>>>(src, dst, cur, agg, N_EDGES);
    fused_gemm2_bias_relu<<<gemmGrid, 256, lds128, stream>>>(agg, Wr[L], cur, Wo[L], br[L],
                                                             nxt, N_NODES, HDIM);
    float* t = cur; cur = nxt; nxt = t;
  }

  // ---- Global mean pool ----
  zero_f32<<<(N_GRAPHS * HDIM + N_GRAPHS + 255) / 256, 256, 0, stream>>>(sums, N_GRAPHS * HDIM + N_GRAPHS);
  pool_accum<<<(int)(((long long)N_NODES * HDIM + 255) / 256), 256, 0, stream>>>(batch, cur, sums, cnt, N_NODES);
  pool_div<<<(N_GRAPHS * HDIM + 255) / 256, 256, 0, stream>>>(sums, cnt, g0);

  // ---- MLP head (4 x 128->128, relu), reusing the WMMA GEMM kernel (M=128) ----
  fused_gemm2_bias_relu<<<1, 256, lds128, stream>>>(g0, W5, nullptr, nullptr, b5, g1, N_GRAPHS, HDIM);
  fused_gemm2_bias_relu<<<1, 256, lds128, stream>>>(g1, W6, nullptr, nullptr, b6, g0, N_GRAPHS, HDIM);
  fused_gemm2_bias_relu<<<1, 256, lds128, stream>>>(g0, W7, nullptr, nullptr, b7, g1, N_GRAPHS, HDIM);
  fused_gemm2_bias_relu<<<1, 256, lds128, stream>>>(g1, W8, nullptr, nullptr, b8, g0, N_GRAPHS, HDIM);

  // ---- Final linear (128x2) + sigmoid ----
  final_sigmoid<<<1, 256, 0, stream>>>(g0, Wl, bl, (float*)d_out);
}